// ReconStep_70300024701430
// MI455X (gfx1250) — compile-verified
//
#include <hip/hip_runtime.h>
#include <stdint.h>

// MLEM step: proj = A*img (COO scatter), ratio = sinos/(proj+eps),
//            bp = A^T*ratio (COO scatter), out = img/effmap*bp.
// Memory-bound: ~805 MB of COO triplet streaming dominates (~35 us floor at
// 23.3 TB/s). Dense arrays (8 MB each) stay resident in the 192 MB L2.
// CDNA5 path: double-buffered GLOBAL_LOAD_ASYNC_TO_LDS_B128 (ASYNCcnt) stages
// the triplet stream into LDS with th:TH_LOAD_NT (don't evict the hot arrays),
// per-thread-private slots (no barriers, only s_wait_asynccnt), consumed via
// ds_load_b128; scatters are no-return agent-scope global_atomic_add_f32.
// WMMA is inapplicable: 0.3 FLOP/byte, no dense tile structure.

typedef float v4f __attribute__((ext_vector_type(4)));
typedef int   v4i __attribute__((ext_vector_type(4)));

#define TPB 256   // 8 wave32 waves per block
#define QPT 8     // target quads (4 nnz) per thread -> ~8 pipelined stages

__device__ __forceinline__ void atomic_add_f32(float* p, float v) {
    // relaxed agent-scope fadd, result unused -> no-return global_atomic_add_f32
    (void)__hip_atomic_fetch_add(p, v, __ATOMIC_RELAXED, __HIP_MEMORY_SCOPE_AGENT);
}

// Low 32 bits of a generic LDS pointer == byte offset within the workgroup's
// LDS allocation (shared aperture lives in addr[63:32]).
__device__ __forceinline__ uint32_t lds_off(const void* p) {
    return (uint32_t)(uintptr_t)p;
}

// One-lane-granular DMA: 16 bytes from this lane's global address to this
// lane's LDS offset. Tracked by ASYNCcnt; completes in issue order.
__device__ __forceinline__ void async_copy_b128(uint32_t lds_byte_off, const void* gptr) {
    asm volatile("global_load_async_to_lds_b128 %0, %1, off th:TH_LOAD_NT"
                 :: "v"(lds_byte_off), "v"(gptr)
                 : "memory");
}

__global__ __launch_bounds__(TPB) void zero_k(float* __restrict__ a, int na,
                                              float* __restrict__ b, int nb) {
    int i = blockIdx.x * TPB + threadIdx.x;
    int stride = gridDim.x * TPB;
    int n = na > nb ? na : nb;
    for (; i < n; i += stride) {
        if (i < na) a[i] = 0.0f;
        if (i < nb) b[i] = 0.0f;
    }
}

// dst[sidx[i]] += vals[i] * src[gidx[i]]  for i in [0, nnz).
// Forward pass: gidx=cols, sidx=rows, src=img, dst=proj.
// Backward pass: gidx=rows, sidx=cols, src=ratio, dst=bp.
__global__ __launch_bounds__(TPB) void spmv_coo_k(
    const float* __restrict__ vals,
    const int*   __restrict__ gidx,
    const int*   __restrict__ sidx,
    const float* __restrict__ src,
    float*       __restrict__ dst,
    int nnz)
{
    // Double-buffered staging: 2 stages x 3 arrays x 256 threads x 16 B = 24 KB
    // of the 320 KB/WGP LDS. Thread t exclusively owns slot [*][*][t].
    __shared__ __align__(16) v4f s_val[2][TPB];
    __shared__ __align__(16) v4i s_g[2][TPB];
    __shared__ __align__(16) v4i s_s[2][TPB];

    const int nquads = nnz >> 2;
    const int tid    = blockIdx.x * TPB + threadIdx.x;
    const int stride = gridDim.x * TPB;
    const int t      = threadIdx.x;

    const uint32_t lv[2] = { lds_off(&s_val[0][t]), lds_off(&s_val[1][t]) };
    const uint32_t lg[2] = { lds_off(&s_g[0][t]),   lds_off(&s_g[1][t])   };
    const uint32_t ls[2] = { lds_off(&s_s[0][t]),   lds_off(&s_s[1][t])   };

    const v4f* __restrict__ v4 = (const v4f*)vals;
    const v4i* __restrict__ g4 = (const v4i*)gidx;
    const v4i* __restrict__ s4 = (const v4i*)sidx;

    // Prologue: stage 0 in flight.
    if (tid < nquads) {
        async_copy_b128(lv[0], v4 + tid);
        async_copy_b128(lg[0], g4 + tid);
        async_copy_b128(ls[0], s4 + tid);
    }

    int buf = 0;
    for (int q = tid; q < nquads; q += stride, buf ^= 1) {
        const int qn = q + stride;
        if (qn < nquads) {
            // Kick off next stage, then wait for the 3 older copies only:
            // async loads decrement ASYNCcnt in order.
            async_copy_b128(lv[buf ^ 1], v4 + qn);
            async_copy_b128(lg[buf ^ 1], g4 + qn);
            async_copy_b128(ls[buf ^ 1], s4 + qn);
            asm volatile("s_wait_asynccnt 0x3" ::: "memory");
        } else {
            asm volatile("s_wait_asynccnt 0x0" ::: "memory");
        }

        // Consume this lane's staged quad from LDS (ds_load_b128).
        v4f v = s_val[buf][t];
        v4i g = s_g[buf][t];
        v4i s = s_s[buf][t];

        // Random gathers from the L2-resident hot array, then fire-and-forget
        // FP32 atomics (random targets over 2M entries -> negligible conflict).
        atomic_add_f32(&dst[s.x], v.x * src[g.x]);
        atomic_add_f32(&dst[s.y], v.y * src[g.y]);
        atomic_add_f32(&dst[s.z], v.z * src[g.z]);
        atomic_add_f32(&dst[s.w], v.w * src[g.w]);
    }

    // Tail (nnz % 4): direct path, handled by the first few global threads.
    const int rem = nnz & 3;
    if (rem && tid < rem) {
        const int idx = (nnz & ~3) + tid;
        atomic_add_f32(&dst[sidx[idx]], vals[idx] * src[gidx[idx]]);
    }
}

// in-place: proj[i] = sinos[i] / (proj[i] + 1e-8f)
__global__ __launch_bounds__(TPB) void ratio_k(float* __restrict__ proj,
                                               const float* __restrict__ sinos,
                                               int n) {
    int i = blockIdx.x * TPB + threadIdx.x;
    int stride = gridDim.x * TPB;
    for (; i < n; i += stride)
        proj[i] = sinos[i] / (proj[i] + 1e-8f);
}

// in-place: out[i] = img[i] / eff[i] * out[i]   (out holds bp on entry)
__global__ __launch_bounds__(TPB) void finish_k(float* __restrict__ out,
                                                const float* __restrict__ img,
                                                const float* __restrict__ eff,
                                                int n) {
    int i = blockIdx.x * TPB + threadIdx.x;
    int stride = gridDim.x * TPB;
    for (; i < n; i += stride)
        out[i] = img[i] / eff[i] * out[i];
}

extern "C" void kernel_launch(void* const* d_in, const int* in_sizes, int n_in,
                              void* d_out, int out_size, void* d_ws, size_t ws_size,
                              hipStream_t stream) {
    // setup_inputs order: img, effmap, sinos, vals, rows, cols, n_lors, n_voxels
    const float* img   = (const float*)d_in[0];
    const float* eff   = (const float*)d_in[1];
    const float* sinos = (const float*)d_in[2];
    const float* vals  = (const float*)d_in[3];
    const int*   rows  = (const int*)d_in[4];
    const int*   cols  = (const int*)d_in[5];

    const int n_vox  = in_sizes[0];   // 2,097,152
    const int n_lors = in_sizes[2];   // 2,000,000
    const int nnz    = in_sizes[3];   // 33,554,432

    float* proj = (float*)d_ws;       // n_lors floats (8 MB) of scratch
    float* out  = (float*)d_out;      // bp accumulator, then final image

    // 1) zero proj and out accumulators
    int nmax = n_lors > n_vox ? n_lors : n_vox;
    int zb = (nmax + TPB - 1) / TPB;
    zero_k<<<zb, TPB, 0, stream>>>(proj, n_lors, out, n_vox);

    // 2) forward projection: proj[rows] += vals * img[cols]
    int nquads = nnz >> 2;
    int blocks = (nquads + TPB * QPT - 1) / (TPB * QPT);
    if (blocks < 1) blocks = 1;
    spmv_coo_k<<<blocks, TPB, 0, stream>>>(vals, cols, rows, img, proj, nnz);

    // 3) ratio = sinos / (proj + 1e-8), in place in proj
    int rb = (n_lors + TPB - 1) / TPB;
    ratio_k<<<rb, TPB, 0, stream>>>(proj, sinos, n_lors);

    // 4) back projection: out[cols] += vals * ratio[rows]
    spmv_coo_k<<<blocks, TPB, 0, stream>>>(vals, rows, cols, proj, out, nnz);

    // 5) out = img / effmap * bp
    int fb = (n_vox + TPB - 1) / TPB;
    finish_k<<<fb, TPB, 0, stream>>>(out, img, eff, n_vox);
}